// TorchDispatchModule_54082228191663
// MI455X (gfx1250) — compile-verified
//
#include <hip/hip_runtime.h>
#include <hip/hip_bf16.h>
#include <stdint.h>

#define NUM_CHIPS   4
#define EPC         8
#define N_EXPERTS   32
#define TOP_K       4
#define SEQ         1024
#define HIDDEN      2048
#define MAX_TOK     1024
#define META_LEN    8
#define N_ASSIGN    (SEQ * TOP_K)            /* 4096 assignments per chip   */
#define TOTAL_SLOTS (N_EXPERTS * MAX_TOK)    /* 32768 slots                 */
#define BUF_ELEMS   ((size_t)TOTAL_SLOTS * HIDDEN)
#define META_ELEMS  ((size_t)TOTAL_SLOTS * META_LEN)

// Native clang vectors: accepted by __builtin_nontemporal_store and lower to
// global_store_b128 th:TH_STORE_NT.
typedef float v4f __attribute__((ext_vector_type(4)));
typedef int   v4i __attribute__((ext_vector_type(4)));

// ---------------- CDNA5 async global<->LDS DMA helpers (ASYNCcnt path) ----
__device__ __forceinline__ void async_load_b128(uint32_t lds_off, const void* gaddr) {
    asm volatile("global_load_async_to_lds_b128 %0, %1, off"
                 :: "v"(lds_off), "v"(gaddr) : "memory");
}
// Non-temporal: buf is written once and never re-read -> don't thrash L2,
// keep the 192MB L2 for the x rows that are re-read ~4x.
__device__ __forceinline__ void async_store_b128_nt(const void* gaddr, uint32_t lds_off) {
    asm volatile("global_store_async_from_lds_b128 %0, %1, off th:TH_STORE_NT"
                 :: "v"(gaddr), "v"(lds_off) : "memory");
}
__device__ __forceinline__ void wait_asynccnt0() {
    asm volatile("s_wait_asynccnt 0" ::: "memory");
}

// ---------------- K0: reset inverse map -----------------------------------
__global__ void moe_init_inv(int* __restrict__ inv) {
    int i = blockIdx.x * blockDim.x + threadIdx.x;
    if (i < TOTAL_SLOTS) inv[i] = -1;
}

// ---------------- K1: per-(chip,expert) counts ----------------------------
__global__ void moe_count(const int* __restrict__ indices, int* __restrict__ counts) {
    int t = threadIdx.x;
    if (t >= NUM_CHIPS * N_EXPERTS) return;
    int c = t >> 5, e = t & 31;
    const int* ind = indices + c * N_ASSIGN;
    int cnt = 0;
    for (int n = 0; n < N_ASSIGN; ++n) cnt += (ind[n] == e);
    counts[c * N_EXPERTS + e] = cnt;
}

// ---------------- K2: build inverse slot map + experts_counter ------------
__global__ void moe_plan(const int* __restrict__ indices,
                         const int* __restrict__ counts,
                         int* __restrict__ inv,
                         int* __restrict__ counter_out) {
    int t = threadIdx.x;
    if (t >= NUM_CHIPS * N_EXPERTS) return;
    int c = t >> 5, e = t & 31;

    // exclusive prefix over chips = this chip's starting offset in expert e's buffer
    int off = 0;
    for (int cc = 0; cc < c; ++cc) off += counts[cc * N_EXPERTS + e];
    int base = e * MAX_TOK + off;

    const int* ind = indices + c * N_ASSIGN;
    int r = 0;
    for (int n = 0; n < N_ASSIGN; ++n) {
        if (ind[n] == e) {
            int flat = base + r;
            if (flat >= 0 && flat < TOTAL_SLOTS)   // JAX scatter drops OOB
                inv[flat] = c * N_ASSIGN + n;      // assignment id
            ++r;
        }
    }
    if (c == 0) {
        int s = 0;
        for (int cc = 0; cc < NUM_CHIPS; ++cc) s += counts[cc * N_EXPERTS + e];
        counter_out[e] = s;
    }
}

// ---------------- K3: slot copy via async LDS DMA + fused metadata --------
// One block per slot (8 KB). Each of the 256 lanes owns 32 bytes of the LDS
// tile: async-load global->LDS, wait, async-store LDS->global (NT). Lanes'
// LDS regions are disjoint, so a per-wave s_wait_asynccnt suffices (no
// barrier). Thread 0 additionally emits the 8-int metadata row.
__global__ void __launch_bounds__(256) moe_copy(const float* __restrict__ x,
                                                const int* __restrict__ inv,
                                                const int* __restrict__ indices,
                                                const float* __restrict__ weights,
                                                float* __restrict__ buf,
                                                int* __restrict__ meta) {
    __shared__ float tile[HIDDEN];    // 8 KB
    int slot = blockIdx.x;
    int t    = threadIdx.x;
    int a    = inv[slot];             // uniform across block
    float* dst = buf + (size_t)slot * HIDDEN + t * 8;

    if (a >= 0) {
        int c   = a >> 12;            // / N_ASSIGN
        int n   = a & (N_ASSIGN - 1);
        int tok = n >> 2;             // / TOP_K
        const float* src = x + ((size_t)(c * SEQ + tok)) * HIDDEN + t * 8;
        // low 32 bits of the flat LDS address == LDS byte offset (aperture map)
        uint32_t lds = (uint32_t)(uintptr_t)(const void*)tile + (uint32_t)t * 32u;

        async_load_b128(lds,       src);
        async_load_b128(lds + 16u, src + 4);

        if (t == 0) {                 // fused metadata for occupied slot
            int k = n & (TOP_K - 1);
            int e = indices[a];
            __hip_bfloat16 bh = __float2bfloat16(weights[a]);  // RNE, matches jnp
            short sb;
            __builtin_memcpy(&sb, &bh, sizeof(sb));
            int wbits = (int)sb;      // sign-extend int16 -> int32 like reference
            v4i* row = (v4i*)(meta + (size_t)slot * META_LEN);
            v4i lo = {c, tok, k, e};
            v4i hi = {wbits, 0, 0, 0};
            __builtin_nontemporal_store(lo, row);
            __builtin_nontemporal_store(hi, row + 1);
        }

        wait_asynccnt0();                       // loads landed in LDS
        async_store_b128_nt(dst,     lds);
        async_store_b128_nt(dst + 4, lds + 16u);
        wait_asynccnt0();                       // drain before LDS realloc
    } else {
        v4f z = {0.f, 0.f, 0.f, 0.f};
        __builtin_nontemporal_store(z, (v4f*)dst);       // global_store_b128 NT
        __builtin_nontemporal_store(z, (v4f*)(dst + 4));
        if (t == 0) {                 // fused metadata default (-1)
            v4i* row = (v4i*)(meta + (size_t)slot * META_LEN);
            v4i m1 = {-1, -1, -1, -1};
            __builtin_nontemporal_store(m1, row);
            __builtin_nontemporal_store(m1, row + 1);
        }
    }
}

// ---------------- launch ---------------------------------------------------
extern "C" void kernel_launch(void* const* d_in, const int* in_sizes, int n_in,
                              void* d_out, int out_size, void* d_ws, size_t ws_size,
                              hipStream_t stream) {
    (void)in_sizes; (void)n_in; (void)out_size; (void)ws_size;

    const float* x       = (const float*)d_in[0];   // [4,1024,2048]
    const float* weights = (const float*)d_in[1];   // [4,1024,4]
    const int*   indices = (const int*)d_in[2];     // [4,1024,4]

    // outputs concatenated flat: buf (f32) | meta (i32) | experts_counter (i32)
    float* buf     = (float*)d_out;
    int*   meta    = (int*)((float*)d_out + BUF_ELEMS);
    int*   counter = meta + META_ELEMS;

    // workspace: counts[4][32] | inv[32768]
    int* counts = (int*)d_ws;
    int* inv    = counts + NUM_CHIPS * N_EXPERTS;

    moe_init_inv<<<TOTAL_SLOTS / 256, 256, 0, stream>>>(inv);
    moe_count   <<<1, 128, 0, stream>>>(indices, counts);
    moe_plan    <<<1, 128, 0, stream>>>(indices, counts, inv, counter);
    moe_copy    <<<TOTAL_SLOTS, 256, 0, stream>>>(x, inv, indices, weights, buf, meta);
}